// HiroLRANDiag_73581379715572
// MI455X (gfx1250) — compile-verified
//
#include <hip/hip_runtime.h>

// ---------------------------------------------------------------------------
// MI455X (gfx1250). All GEMMs on V_WMMA_F32_16X16X4_F32 (f32, wave32).
// Problem is memory/latency bound (~130MB @ 23.3TB/s, ~34 GFLOP), so full
// f32 precision on the matrix pipe is the right operating point.
// Data movement: GLOBAL_LOAD_ASYNC_TO_LDS (ASYNCcnt) for weight panels
// (transposed scatter + double buffering), activations, and output.
// ---------------------------------------------------------------------------

typedef __attribute__((ext_vector_type(2))) float v2f;
typedef __attribute__((ext_vector_type(8))) float v8f;
typedef __attribute__((ext_vector_type(4))) int   v4i;

#define BATCH   32
#define SEQ     2048
#define STATE   64
#define LATENT  128
#define INPUT_DIM 320
#define TOKENS  (BATCH * SEQ)      // 65536
#define CHUNK   64
#define NCHUNK  (SEQ / CHUNK)      // 32

// LDS strides padded against bank conflicts; ACT_S also 16B-aligned rows.
#define ACT_S   260   // activation rows (f32): 1040B row, (4*lane)%64 banks
#define WPAN_S  66    // transposed weight panel [n][k], k<64
#define AT_S    130   // transposed A matrix [n][k]
#define S_S     130   // scan state rows

typedef __attribute__((address_space(1))) int g_int;   // global (AS1)
typedef __attribute__((address_space(3))) int l_int;   // LDS (AS3)
typedef __attribute__((address_space(1))) v4i g_v4i;
typedef __attribute__((address_space(3))) v4i l_v4i;

__device__ __forceinline__ v8f wmma_f32(v2f a, v2f b, v8f c) {
  return __builtin_amdgcn_wmma_f32_16x16x4_f32(
      false, a, false, b, (short)0, c, false, false);
}

// ---- async global->LDS / LDS->global helpers (ASYNCcnt-tracked) -----------
__device__ __forceinline__ void async_ld_b32(const float* gsrc, float* ldst) {
#if __has_builtin(__builtin_amdgcn_global_load_async_to_lds_b32)
  __builtin_amdgcn_global_load_async_to_lds_b32(
      (g_int*)gsrc, (l_int*)ldst, 0, 0);
#else
  unsigned l = (unsigned)(unsigned long long)(l_int*)ldst;
  asm volatile("global_load_async_to_lds_b32 %0, %1, off"
               :: "v"(l), "v"(gsrc) : "memory");
#endif
}
__device__ __forceinline__ void async_ld_b128(const float* gsrc, float* ldst) {
#if __has_builtin(__builtin_amdgcn_global_load_async_to_lds_b128)
  __builtin_amdgcn_global_load_async_to_lds_b128(
      (g_v4i*)gsrc, (l_v4i*)ldst, 0, 0);
#else
  unsigned l = (unsigned)(unsigned long long)(l_int*)ldst;
  asm volatile("global_load_async_to_lds_b128 %0, %1, off"
               :: "v"(l), "v"(gsrc) : "memory");
#endif
}
__device__ __forceinline__ void async_st_b128(float* gdst, const float* lsrc) {
#if __has_builtin(__builtin_amdgcn_global_store_async_from_lds_b128)
  __builtin_amdgcn_global_store_async_from_lds_b128(
      (g_v4i*)gdst, (l_v4i*)lsrc, 0, 0);
#else
  unsigned l = (unsigned)(unsigned long long)(l_int*)lsrc;
  asm volatile("global_store_async_from_lds_b128 %0, %1, off"
               :: "v"(gdst), "v"(l) : "memory");
#endif
}
__device__ __forceinline__ void wait_async0() {
#if __has_builtin(__builtin_amdgcn_s_wait_asynccnt)
  __builtin_amdgcn_s_wait_asynccnt(0);
#else
  asm volatile("s_wait_asynccnt 0" ::: "memory");
#endif
}

// stage transposed weight panel asynchronously: wpan[n][k] = W[kb+k][n]
template <int N>
__device__ __forceinline__ void stage_panel(const float* __restrict__ Wg,
                                            float* wpan, int kb, int tid) {
  for (int idx = tid; idx < 64 * N; idx += 512) {
    int k = idx / N, n = idx - k * N;
    async_ld_b32(Wg + (size_t)(kb + k) * N + n, wpan + n * WPAN_S + k);
  }
}

// ---------------------------------------------------------------------------
// Fused MLP layer with double-buffered async weight panels.
// 16 waves = (4 token row-tiles) x (4 column groups).
// ---------------------------------------------------------------------------
template <int K, int N, bool RELU>
__device__ __forceinline__ void mlp_layer(const float* __restrict__ Wg,
                                          const float* __restrict__ Bg,
                                          const float* actIn, float* actOut,
                                          float* wpan0, float* wpan1) {
  const int tid  = threadIdx.x;
  const int wave = tid >> 5;
  const int lane = tid & 31;
  const int r    = wave & 3;
  const int nq   = wave >> 2;
  const int l15  = lane & 15;
  const int lh   = lane >> 4;
  constexpr int NTP = (N / 16) / 4;

  v8f acc[NTP] = {};
  stage_panel<N>(Wg, wpan0, 0, tid);
  float* wbuf[2] = {wpan0, wpan1};
  int pb = 0;

  for (int kb = 0; kb < K; kb += 64) {
    wait_async0();        // this wave's panel (and prior staging) landed
    __syncthreads();      // ... and everyone else's
    if (kb + 64 < K) stage_panel<N>(Wg, wbuf[pb ^ 1], kb + 64, tid);
    const float* wcur = wbuf[pb];
    const float* arow = &actIn[(r * 16 + l15) * ACT_S + kb];
#pragma unroll
    for (int kt = 0; kt < 16; ++kt) {
      int kk = kt * 4 + lh * 2;
      v2f a = *(const v2f*)&arow[kk];
#pragma unroll
      for (int q = 0; q < NTP; ++q) {
        int n = (nq * NTP + q) * 16 + l15;
        v2f b = *(const v2f*)&wcur[n * WPAN_S + kk];
        acc[q] = wmma_f32(a, b, acc[q]);
      }
    }
    pb ^= 1;
  }
  __syncthreads();
#pragma unroll
  for (int q = 0; q < NTP; ++q) {
    int n = (nq * NTP + q) * 16 + l15;
    float bias = Bg[n];
#pragma unroll
    for (int e = 0; e < 8; ++e) {
      int m = lh * 8 + e;
      float v = acc[q][e] + bias;
      if (RELU) v = fmaxf(v, 0.0f);
      actOut[(r * 16 + m) * ACT_S + n] = v;
    }
  }
  __syncthreads();
}

// ---------------------------------------------------------------------------
// Encoder: x(64)->256->256->z(128); BN sum/sumsq in registers + one atomic;
// stores z[:,0,:] (the only normalized slice the scan consumes).
// ---------------------------------------------------------------------------
__global__ void __launch_bounds__(512)
enc_kernel(const float* __restrict__ in, const float* __restrict__ w0,
           const float* __restrict__ b0, const float* __restrict__ w1,
           const float* __restrict__ b1, const float* __restrict__ w2,
           const float* __restrict__ b2, float* __restrict__ bn_sum,
           float* __restrict__ bn_sq, float* __restrict__ z0) {
  extern __shared__ char smem[];
  float* actA  = (float*)smem;               // 64 * ACT_S
  float* actB  = actA + 64 * ACT_S;
  float* wpan0 = actB + 64 * ACT_S;          // 256 * WPAN_S
  float* wpan1 = wpan0 + 256 * WPAN_S;
  const int tid = threadIdx.x;
  float rsum = 0.0f, rsq = 0.0f;

  for (int mt = blockIdx.x; mt < TOKENS / 64; mt += gridDim.x) {
    int base = mt * 64;
    // async stage x slice: 64 rows x 16 f32 = 4 b128 chunks per row
    for (int idx = tid; idx < 64 * 4; idx += 512) {
      int row = idx >> 2, c4 = idx & 3;
      async_ld_b128(in + (size_t)(base + row) * INPUT_DIM + c4 * 4,
                    actA + row * ACT_S + c4 * 4);
    }
    mlp_layer<64,  256, true>(w0, b0, actA, actB, wpan0, wpan1);
    mlp_layer<256, 256, true>(w1, b1, actB, actA, wpan0, wpan1);
    mlp_layer<256, 128, true>(w2, b2, actA, actB, wpan0, wpan1);
    if (tid < 128) {
      for (int row = 0; row < 64; ++row) {
        float v = actB[row * ACT_S + tid];
        rsum += v;
        rsq  += v * v;
        int token = base + row;
        if ((token & (SEQ - 1)) == 0)
          z0[(token >> 11) * LATENT + tid] = v;
      }
    }
    __syncthreads();
  }
  if (tid < 128) {
    atomicAdd(&bn_sum[tid], rsum);
    atomicAdd(&bn_sq[tid], rsq);
  }
}

// ---------------------------------------------------------------------------
__global__ void bn_kernel(const float* __restrict__ bn_sum,
                          const float* __restrict__ bn_sq,
                          const float* __restrict__ gamma,
                          const float* __restrict__ beta,
                          const float* __restrict__ z0,
                          float* __restrict__ carry0) {
  int n = threadIdx.x;  // 128
  float mean  = bn_sum[n] * (1.0f / TOKENS);
  float var   = bn_sq[n] * (1.0f / TOKENS) - mean * mean;
  float scale = rsqrtf(var + 1e-5f) * gamma[n];
  float sh    = beta[n];
  for (int b = 0; b < BATCH; ++b)
    carry0[b * LATENT + n] = (z0[b * LATENT + n] - mean) * scale + sh;
}

// ---------------------------------------------------------------------------
// Chunked scan. Per step/wave: 32 chained f32 WMMAs (K=128).
// B-fragments (A^T rows) hoisted into 32 v2f registers (loop-invariant).
// u_t loads software-pipelined one step ahead (hidden under the WMMA chain).
// ---------------------------------------------------------------------------
template <bool EMIT>
__global__ void __launch_bounds__(512)
scan_kernel(const float* __restrict__ in, const float* __restrict__ Aw,
            const float* __restrict__ Bdiag, const float* __restrict__ carry,
            float* __restrict__ Eout, float* __restrict__ z1) {
  extern __shared__ char smem[];
  float* At = (float*)smem;            // 128 * AT_S  (A^T: [n][k])
  float* Sa = At + 128 * AT_S;
  float* Sb = Sa + 32 * S_S;
  const int tid  = threadIdx.x;
  const int wave = tid >> 5, lane = tid & 31;
  const int l15  = lane & 15, lh = lane >> 4;
  const int mi   = wave & 1, ni = wave >> 1;
  const int c    = blockIdx.x;

  // async transposed stage of A
  for (int idx = tid; idx < 128 * 128; idx += 512) {
    int k = idx >> 7, n = idx & 127;
    async_ld_b32(Aw + idx, At + n * AT_S + k);
  }
  for (int idx = tid; idx < BATCH * LATENT; idx += 512) {
    int m = idx >> 7, n = idx & 127;
    float v;
    if (EMIT) v = carry[(size_t)c * BATCH * LATENT + idx];
    else      v = (c == 0) ? carry[idx] : 0.0f;
    Sa[m * S_S + n] = v;
  }
  wait_async0();
  __syncthreads();

  const int nlane = ni * 16 + l15;
  float d = Bdiag[nlane];
  d = fminf(fmaxf(d, -0.95f), 0.95f);

  // hoist loop-invariant B-fragments (64 VGPRs)
  v2f bfrag[32];
#pragma unroll
  for (int kt = 0; kt < 32; ++kt)
    bfrag[kt] = *(const v2f*)&At[nlane * AT_S + kt * 4 + lh * 2];

  // pipeline u one step ahead
  float u[8];
#pragma unroll
  for (int e = 0; e < 8; ++e) {
    int m = mi * 16 + lh * 8 + e;
    u[e] = in[((size_t)m * SEQ + c * CHUNK) * INPUT_DIM + (STATE + LATENT) + nlane];
  }

  float* Scur = Sa;
  float* Snxt = Sb;
  const int arow = (mi * 16 + l15) * S_S;
  for (int s = 0; s < CHUNK; ++s) {
    int t = c * CHUNK + s;
    v8f acc = {};
#pragma unroll
    for (int kt = 0; kt < 32; ++kt) {
      v2f a = *(const v2f*)&Scur[arow + kt * 4 + lh * 2];
      acc = wmma_f32(a, bfrag[kt], acc);
    }
#pragma unroll
    for (int e = 0; e < 8; ++e) {
      int m = mi * 16 + lh * 8 + e;  // batch index
      float v = acc[e] + u[e] * d;
      Snxt[m * S_S + nlane] = v;
      if (EMIT) z1[((size_t)m * SEQ + t) * LATENT + nlane] = v;
    }
    if (s + 1 < CHUNK) {
#pragma unroll
      for (int e = 0; e < 8; ++e) {
        int m = mi * 16 + lh * 8 + e;
        u[e] = in[((size_t)m * SEQ + t + 1) * INPUT_DIM + (STATE + LATENT) + nlane];
      }
    }
    __syncthreads();
    float* tmp = Scur; Scur = Snxt; Snxt = tmp;
  }
  if (!EMIT) {
    for (int idx = tid; idx < BATCH * LATENT; idx += 512) {
      int m = idx >> 7, n = idx & 127;
      Eout[(size_t)c * BATCH * LATENT + idx] = Scur[m * S_S + n];
    }
  }
}

// ---------------------------------------------------------------------------
// Carry combine: A^64 via 6 squarings, then serial carry recurrence.
// ---------------------------------------------------------------------------
__global__ void __launch_bounds__(512)
combine_kernel(const float* __restrict__ Aw, float* __restrict__ carry,
               const float* __restrict__ E) {
  extern __shared__ char smem[];
  float* X  = (float*)smem;          // [m][k]
  float* XT = X  + 128 * AT_S;       // [n][k]
  float* Y  = XT + 128 * AT_S;
  float* YT = Y  + 128 * AT_S;
  float* Ca = YT + 128 * AT_S;
  float* Cb = Ca + 32 * S_S;
  const int tid  = threadIdx.x;
  const int wave = tid >> 5, lane = tid & 31;
  const int l15  = lane & 15, lh = lane >> 4;

  for (int idx = tid; idx < 128 * 128; idx += 512) {
    int k = idx >> 7, n = idx & 127;   // element A[k][n]
    X [k * AT_S + n] = Aw[idx];
    XT[n * AT_S + k] = Aw[idx];
  }
  __syncthreads();

  for (int sq = 0; sq < 6; ++sq) {
#pragma unroll
    for (int j = 0; j < 4; ++j) {
      int tile = wave * 4 + j;
      int tmi = tile >> 3, tni = tile & 7;
      v8f acc = {};
      for (int kt = 0; kt < 32; ++kt) {
        int kk = kt * 4 + lh * 2;
        v2f a = *(const v2f*)&X [(tmi * 16 + l15) * AT_S + kk];
        v2f b = *(const v2f*)&XT[(tni * 16 + l15) * AT_S + kk];
        acc = wmma_f32(a, b, acc);
      }
#pragma unroll
      for (int e = 0; e < 8; ++e) {
        int m = tmi * 16 + lh * 8 + e, n = tni * 16 + l15;
        Y [m * AT_S + n] = acc[e];
        YT[n * AT_S + m] = acc[e];
      }
    }
    __syncthreads();
    float* t;
    t = X;  X  = Y;  Y  = t;
    t = XT; XT = YT; YT = t;
  }

  const int mi = wave & 1, ni = wave >> 1;
  const int nlane = ni * 16 + l15;
  // hoist (A^64)^T fragments — invariant across the 30 serial carry GEMMs
  v2f xb[32];
#pragma unroll
  for (int kt = 0; kt < 32; ++kt)
    xb[kt] = *(const v2f*)&XT[nlane * AT_S + kt * 4 + lh * 2];

  for (int idx = tid; idx < BATCH * LATENT; idx += 512) {
    int m = idx >> 7, n = idx & 127;
    float v = E[idx];
    carry[BATCH * LATENT + idx] = v;   // carry[1] = E[0]
    Ca[m * S_S + n] = v;
  }
  __syncthreads();

  for (int cc = 1; cc <= 30; ++cc) {
    v8f acc = {};
#pragma unroll
    for (int kt = 0; kt < 32; ++kt) {
      v2f a = *(const v2f*)&Ca[(mi * 16 + l15) * S_S + kt * 4 + lh * 2];
      acc = wmma_f32(a, xb[kt], acc);
    }
#pragma unroll
    for (int e = 0; e < 8; ++e) {
      int m = mi * 16 + lh * 8 + e;
      float v = acc[e] + E[(size_t)cc * BATCH * LATENT + m * LATENT + nlane];
      carry[(size_t)(cc + 1) * BATCH * LATENT + m * LATENT + nlane] = v;
      Cb[m * S_S + nlane] = v;
    }
    __syncthreads();
    float* t = Ca; Ca = Cb; Cb = t;
  }
}

// ---------------------------------------------------------------------------
// Decoder: z1(128)->256->256->out(64); async in/out staging.
// ---------------------------------------------------------------------------
__global__ void __launch_bounds__(512)
dec_kernel(const float* __restrict__ z1, const float* __restrict__ w0,
           const float* __restrict__ b0, const float* __restrict__ w1,
           const float* __restrict__ b1, const float* __restrict__ w2,
           const float* __restrict__ b2, float* __restrict__ out) {
  extern __shared__ char smem[];
  float* actA  = (float*)smem;
  float* actB  = actA + 64 * ACT_S;
  float* wpan0 = actB + 64 * ACT_S;
  float* wpan1 = wpan0 + 256 * WPAN_S;
  const int tid = threadIdx.x;

  for (int mt = blockIdx.x; mt < TOKENS / 64; mt += gridDim.x) {
    int base = mt * 64;
    // async stage z1: 64 rows x 32 f32 = 8 b128 chunks per row
    for (int idx = tid; idx < 64 * 8; idx += 512) {
      int row = idx >> 3, c4 = idx & 7;
      async_ld_b128(z1 + (size_t)(base + row) * LATENT + c4 * 4,
                    actA + row * ACT_S + c4 * 4);
    }
    mlp_layer<128, 256, true >(w0, b0, actA, actB, wpan0, wpan1);
    mlp_layer<256, 256, true >(w1, b1, actB, actA, wpan0, wpan1);
    mlp_layer<256, 64,  false>(w2, b2, actA, actB, wpan0, wpan1);
    // async store output rows (drained by next layer0's wait / s_endpgm)
    for (int idx = tid; idx < 64 * 4; idx += 512) {
      int row = idx >> 2, c4 = idx & 3;
      async_st_b128(out + (size_t)(base + row) * STATE + c4 * 4,
                    actB + row * ACT_S + c4 * 4);
    }
    __syncthreads();
  }
}

// ---------------------------------------------------------------------------
extern "C" void kernel_launch(void* const* d_in, const int* in_sizes, int n_in,
                              void* d_out, int out_size, void* d_ws, size_t ws_size,
                              hipStream_t stream) {
  (void)in_sizes; (void)n_in; (void)out_size; (void)ws_size;
  const float* in    = (const float*)d_in[0];
  const float* ew0   = (const float*)d_in[1];
  const float* eb0   = (const float*)d_in[2];
  const float* ew1   = (const float*)d_in[3];
  const float* eb1   = (const float*)d_in[4];
  const float* ew2   = (const float*)d_in[5];
  const float* eb2   = (const float*)d_in[6];
  const float* gamma = (const float*)d_in[7];
  const float* beta  = (const float*)d_in[8];
  const float* Aw    = (const float*)d_in[9];
  const float* Bd    = (const float*)d_in[10];
  const float* dw0   = (const float*)d_in[11];
  const float* db0   = (const float*)d_in[12];
  const float* dw1   = (const float*)d_in[13];
  const float* db1   = (const float*)d_in[14];
  const float* dw2   = (const float*)d_in[15];
  const float* db2   = (const float*)d_in[16];

  float* ws     = (float*)d_ws;
  float* bn_sum = ws;                          // 128
  float* bn_sq  = ws + 128;                    // 128
  float* z0     = ws + 256;                    // 32*128
  float* carry  = z0 + BATCH * LATENT;         // NCHUNK * 32*128
  float* E      = carry + NCHUNK * BATCH * LATENT;
  float* z1     = E + NCHUNK * BATCH * LATENT; // 65536*128 f32

  (void)hipMemsetAsync(bn_sum, 0, 256 * sizeof(float), stream);

  const size_t mlp_lds  = (size_t)(2 * 64 * ACT_S + 2 * 256 * WPAN_S) * sizeof(float);
  const size_t scan_lds = (size_t)(128 * AT_S + 2 * 32 * S_S) * sizeof(float);
  const size_t comb_lds = (size_t)(4 * 128 * AT_S + 2 * 32 * S_S) * sizeof(float);

  enc_kernel<<<256, 512, mlp_lds, stream>>>(in, ew0, eb0, ew1, eb1, ew2, eb2,
                                            bn_sum, bn_sq, z0);
  bn_kernel<<<1, 128, 0, stream>>>(bn_sum, bn_sq, gamma, beta, z0, carry);
  scan_kernel<false><<<NCHUNK, 512, scan_lds, stream>>>(in, Aw, Bd, carry, E, nullptr);
  combine_kernel<<<1, 512, comb_lds, stream>>>(Aw, carry, E);
  scan_kernel<true><<<NCHUNK, 512, scan_lds, stream>>>(in, Aw, Bd, carry, nullptr, z1);
  dec_kernel<<<256, 512, mlp_lds, stream>>>(z1, dw0, db0, dw1, db1, dw2, db2,
                                            (float*)d_out);
}